// UpSample2d_87290915324550
// MI455X (gfx1250) — compile-verified
//
#include <hip/hip_runtime.h>
#include <hip/hip_bf16.h>

// ---------------------------------------------------------------------------
// 2x Kaiser-sinc upsample (depthwise conv_transpose2d, k=12, stride=2,
// replicate pad) for x:(4,64,256,256) f32 -> out:(4,64,512,512) f32.
//
// Per-axis math (derived from the reference's pad/dilate/crop algebra):
//   n even, r=n/2    : out[n] = 4 * sum_{a=0..5} filt[11-2a] * x[clamp(r-3+a)]
//   n odd,  r=(n-1)/2: out[n] = 4 * sum_{a=0..5} filt[10-2a] * x[clamp(r-2+a)]
// 2D: out[2r+py,2c+px] = sum_{ay,ax} W[py,px][ay,ax]*xin[r+py+ay, c+px+ax]
// with xin = clamp-padded input (tile base r0-3, c0-3),
//      W[py,px][ay,ax] = 4*filt[Spy(ay)*12 + Spx(ax)], S0(a)=11-2a, S1(a)=10-2a.
//
// WMMA (banded-matrix) formulation per 16x16 phase tile:
//   D(16x16) += sum_{ay=0..5} sum_{kc=0..5} A(ay,kc) x B(ay,kc)
//   A[m,k] = xin[tileY*16+m+py+ay, tileX*16+4kc+k]      (16x4 f32, K=4)
//   B[k,n] = (0<=k-n-px<=5) ? W[py,px][ay,k-n-px] : 0   (banded 4x16)
// Full fp32 via V_WMMA_F32_16X16X4_F32 (RNE) -> reference-class accuracy.
//
// Round-2 feedback applied: the allocator single-buffered A fragments
// (ds_load -> s_wait_dscnt 0x0 -> 4 wmma, serialized). Now each ay-row's six
// A fragments are loaded into distinct registers first (3 ds_load_2addr_b64),
// then the 12-WMMA chain runs, so DS completion overlaps WMMA issue and the
// waits stagger (dscnt 2/1/0) instead of hitting 0 every 4 WMMAs.
// ---------------------------------------------------------------------------

#define USE_ASYNC_LDS 1

typedef float v2f __attribute__((ext_vector_type(2)));
typedef float v8f __attribute__((ext_vector_type(8)));

#define H_IN 256
#define W_IN 256
// Workgroup output region: 64 (y) x 512 (x) -> phase-plane region 32 x 256.
// Input halo tile: rows 38, cols 262 (= 256 + 6), stride 264 (keeps b64
// alignment; the two pad columns are only ever multiplied by zero B entries).
#define TROWS   38
#define TCOLS   262
#define TSTRIDE 264
#define TILES_X_PER_WAVE 8

// Phase-coefficient lookup: coefficient for horizontal phase p, tap ax.
__device__ __forceinline__ float coefv(const float* __restrict__ f,
                                       int rowb, int p, int ax) {
  if (ax < 0 || ax > 5) return 0.0f;
  const int sx = p ? (10 - 2 * ax) : (11 - 2 * ax);
  return 4.0f * f[rowb + sx];
}

__global__ __launch_bounds__(256)
void upsample2d_wmma(const float* __restrict__ x,
                     const float* __restrict__ filt,
                     float* __restrict__ out) {
  const int tid = threadIdx.x;
  const int img = blockIdx.z;          // fused (b, c) image index
  const int r0  = blockIdx.y * 32;     // phase-plane row base of this WG
  const int c0  = blockIdx.x * 256;    // phase-plane col base of this WG

  __shared__ float tile[TROWS * TSTRIDE];   // 40128 B

  const float* xImg = x + (size_t)img * (H_IN * W_IN);

  // ---- Stage clamped input halo tile (38 x 262) into LDS -----------------
  // Per-lane clamped global gather -> LDS via CDNA5 async load-to-LDS.
  for (int idx = tid; idx < TROWS * TCOLS; idx += 256) {
    const int rr = idx / TCOLS;
    const int cc = idx - rr * TCOLS;
    int gy = r0 - 3 + rr; gy = gy < 0 ? 0 : (gy > H_IN - 1 ? H_IN - 1 : gy);
    int gx = c0 - 3 + cc; gx = gx < 0 ? 0 : (gx > W_IN - 1 ? W_IN - 1 : gx);
    const float* gp = xImg + gy * W_IN + gx;
#if USE_ASYNC_LDS
    // Generic pointers to LDS carry the LDS offset in their low 32 bits.
    unsigned lp = (unsigned)(uintptr_t)&tile[rr * TSTRIDE + cc];
    asm volatile("global_load_async_to_lds_b32 %0, %1, off"
                 :: "v"(lp), "v"(gp) : "memory");
#else
    tile[rr * TSTRIDE + cc] = *gp;
#endif
  }
#if USE_ASYNC_LDS
  asm volatile("s_wait_asynccnt 0" ::: "memory");
#endif
  __syncthreads();

  // ---- Wave task: (py, tileY, tileX-half); px=0/1 both done per wave -----
  const int lane = tid & 31;
  const int w    = tid >> 5;     // 8 waves
  const int lm   = lane & 15;    // A-matrix row / D-matrix column
  const int half = lane >> 4;    // A: K pair select; D: row +8 select
  const int py    = w & 1;
  const int tileY = (w >> 1) & 1;
  const int txp   = (w >> 2) & 1;

  // ---- Build banded B fragments in registers (per ISA B layout:
  //      VGPR0 = rows K=0 (lanes 0-15) / K=2 (lanes 16-31); VGPR1 = K=1/K=3)
  v2f B0[36], B1[36];
#pragma unroll
  for (int ay = 0; ay < 6; ++ay) {
    const int sy   = py ? (10 - 2 * ay) : (11 - 2 * ay);
    const int rowb = sy * 12;
#pragma unroll
    for (int kc = 0; kc < 6; ++kc) {
      const int k0 = 4 * kc + 2 * half;
      v2f b0, b1;
      b0.x = coefv(filt, rowb, 0, k0     - lm);
      b0.y = coefv(filt, rowb, 0, k0 + 1 - lm);
      b1.x = coefv(filt, rowb, 1, k0     - lm - 1);
      b1.y = coefv(filt, rowb, 1, k0 + 1 - lm - 1);
      B0[ay * 6 + kc] = b0;
      B1[ay * 6 + kc] = b1;
    }
  }

  float* outImg = out + (size_t)img * (2 * H_IN) * (2 * W_IN);

#pragma unroll 1
  for (int t = 0; t < TILES_X_PER_WAVE; ++t) {
    const int tileX = txp * TILES_X_PER_WAVE + t;
    v8f D0 = {};   // px = 0 accumulator
    v8f D1 = {};   // px = 1 accumulator
#pragma unroll
    for (int ay = 0; ay < 6; ++ay) {
      // A fragments per ISA 16x4 f32 layout: lane reads b64 of
      // xin[tileY*16 + (lane%16) + py + ay,  tileX*16 + 4kc + 2*(lane/16)]
      const int rbase = (tileY * 16 + lm + py + ay) * TSTRIDE
                      + tileX * 16 + 2 * half;
      // Load the whole ay-row of A fragments first (3x ds_load_2addr_b64
      // into distinct regs) so DS completion overlaps the WMMA chain.
      v2f a[6];
#pragma unroll
      for (int kc = 0; kc < 6; ++kc)
        a[kc] = *(const v2f*)&tile[rbase + 4 * kc];
#pragma unroll
      for (int kc = 0; kc < 6; ++kc) {
        D0 = __builtin_amdgcn_wmma_f32_16x16x4_f32(
                 false, a[kc], false, B0[ay * 6 + kc], (short)0, D0, false, false);
        D1 = __builtin_amdgcn_wmma_f32_16x16x4_f32(
                 false, a[kc], false, B1[ay * 6 + kc], (short)0, D1, false, false);
      }
    }
    // ---- Store: lane holds D[v + 8*half, lm]; pair px=0/1 -> b64 stores --
#pragma unroll
    for (int v = 0; v < 8; ++v) {
      const int m  = v + 8 * half;
      const int ny = 2 * (r0 + tileY * 16 + m) + py;
      const int nx = 2 * (c0 + tileX * 16 + lm);
      v2f val; val.x = D0[v]; val.y = D1[v];
      *(v2f*)(outImg + (size_t)ny * (2 * W_IN) + nx) = val;
    }
  }
}

extern "C" void kernel_launch(void* const* d_in, const int* in_sizes, int n_in,
                              void* d_out, int out_size, void* d_ws, size_t ws_size,
                              hipStream_t stream) {
  const float* x    = (const float*)d_in[0];
  const float* filt = (const float*)d_in[1];   // 1x1x12x12 fp32
  float* out        = (float*)d_out;

  const int BC = in_sizes[0] / (H_IN * W_IN);  // 4*64 = 256 images
  // Each WG: 64 (y) x 512 (x) outputs. Output is 512 x 512 per image.
  dim3 grid((2 * W_IN) / 512, (2 * H_IN) / 64, BC);
  upsample2d_wmma<<<grid, 256, 0, stream>>>(x, filt, out);

  (void)n_in; (void)out_size; (void)d_ws; (void)ws_size;
}